// GCN_13494787244283
// MI455X (gfx1250) — compile-verified
//
#include <hip/hip_runtime.h>

// GCN 2-layer forward for MI455X (gfx1250, wave32).
// Bottleneck is the edge scatter/gather (~3.3 GB irregular traffic/layer @ 23.3 TB/s);
// the 128x128 GEMMs (6.6 GFLOP total) run on V_WMMA_F32_16X16X4_F32 in fp32.

typedef __attribute__((ext_vector_type(2))) float v2f;
typedef __attribute__((ext_vector_type(8))) float v8f;

#define HID   128
#define TROWS 64            // rows of the activation matrix per workgroup
#define LDSP  (HID + 4)     // padded LDS row stride (floats) -> conflict-free ds reads

// ---------------------------------------------------------------- utilities
// vectorized zero: one float4 per thread
__global__ void zero_f32v4(float4* __restrict__ p, int n4) {
    int i = blockIdx.x * blockDim.x + threadIdx.x;
    if (i < n4) p[i] = make_float4(0.f, 0.f, 0.f, 0.f);
}

// exact integer degree counts
__global__ void degree_count(const int* __restrict__ src, const int* __restrict__ dst,
                             unsigned* __restrict__ degO, unsigned* __restrict__ degI, int E) {
    int e = blockIdx.x * blockDim.x + threadIdx.x;
    if (e < E) {
        atomicAdd(&degO[src[e]], 1u);
        atomicAdd(&degI[dst[e]], 1u);
    }
}

// in-place uint count -> rsqrt(max(deg,1)) as float
__global__ void deg_to_norm(float* __restrict__ buf, int n) {
    int i = blockIdx.x * blockDim.x + threadIdx.x;
    if (i < n) {
        unsigned d = ((const unsigned*)buf)[i];
        float fd = (d < 1u) ? 1.0f : (float)d;
        buf[i] = rsqrtf(fd);
    }
}

// ------------------------------------------- t = (h * out_norm[:,None]) @ W
// 128 threads = 4 waves; each wave computes a 16x128 output tile with
// 8 col-tiles x 32 K-steps of V_WMMA_F32_16X16X4_F32.
__global__ __launch_bounds__(128) void gcn_transform_wmma(
    const float* __restrict__ h, const float* __restrict__ norm,
    const float* __restrict__ W, float* __restrict__ t, int N) {

    __shared__ float As[TROWS * LDSP];

    const int rowBase  = blockIdx.x * TROWS;
    const int tid      = threadIdx.x;                 // == column index during staging
    const bool full    = (rowBase + TROWS) <= N;      // 1562 of 1563 blocks

    // Stage 64x128 tile, pre-scaled by out_norm. Coalesced 512B rows.
    if (full) {
#pragma unroll 4
        for (int r = 0; r < TROWS; ++r) {
            int row = rowBase + r;
            As[r * LDSP + tid] = h[(size_t)row * HID + tid] * norm[row];
        }
    } else {
        for (int r = 0; r < TROWS; ++r) {
            int row = rowBase + r;
            float v = 0.0f;
            if (row < N) v = h[(size_t)row * HID + tid] * norm[row];
            As[r * LDSP + tid] = v;
        }
    }
    __syncthreads();

    const int wave = tid >> 5;
    const int lane = tid & 31;
    const int m    = lane & 15;           // M (and N) index within a 16-tile
    const int kb   = (lane >> 4) << 1;    // lanes 0-15: K+{0,1}; lanes 16-31: K+{2,3}
    const int rl   = wave * 16 + m;       // local row for A fragment

    v8f acc[8];
    const v8f vzero = {};
#pragma unroll
    for (int nt = 0; nt < 8; ++nt) acc[nt] = vzero;

    for (int kt = 0; kt < 32; ++kt) {
        const int k = kt * 4 + kb;
        v2f a;
        a.x = As[rl * LDSP + k];          // ds_load_2addr pair; +4 pad spreads banks
        a.y = As[rl * LDSP + k + 1];
#pragma unroll
        for (int nt = 0; nt < 8; ++nt) {
            const int n = nt * 16 + m;
            v2f b;
            b.x = W[k * HID + n];         // 64KB weight stays hot in WGP$/L2
            b.y = W[(k + 1) * HID + n];
            acc[nt] = __builtin_amdgcn_wmma_f32_16x16x4_f32(
                false, a, false, b, (short)0, acc[nt], false, false);
        }
    }

    // C/D layout: VGPR r -> M=r (lanes 0-15) / M=8+r (lanes 16-31), N=lane&15
    const int rowHalf = (lane >> 4) * 8;
    const int rowW    = rowBase + wave * 16 + rowHalf;
    if (full) {
#pragma unroll
        for (int nt = 0; nt < 8; ++nt) {
            const int col = nt * 16 + m;
#pragma unroll
            for (int r = 0; r < 8; ++r)
                t[(size_t)(rowW + r) * HID + col] = acc[nt][r];
        }
    } else {
#pragma unroll
        for (int nt = 0; nt < 8; ++nt) {
            const int col = nt * 16 + m;
#pragma unroll
            for (int r = 0; r < 8; ++r)
                if (rowW + r < N) t[(size_t)(rowW + r) * HID + col] = acc[nt][r];
        }
    }
}

// ------------------------------------------------ agg[dst] += t[src] (edges)
// 128 consecutive threads own one edge: 512B coalesced gather + 512B of
// hardware global_atomic_add_f32 (unsafeAtomicAdd avoids the CAS loop).
__global__ __launch_bounds__(256) void gcn_scatter(
    const float* __restrict__ t, const int* __restrict__ src,
    const int* __restrict__ dst, float* __restrict__ agg, int E) {
    int gid = blockIdx.x * 256 + threadIdx.x;
    int e = gid >> 7;
    int f = gid & 127;
    if (e < E) {
        int s = src[e];
        int d = dst[e];
        float v = t[(size_t)s * HID + f];
        unsafeAtomicAdd(&agg[(size_t)d * HID + f], v);
    }
}

// --------------------------------------- out = relu(agg * in_norm[:,None]+b)
// one float4 per thread: 32 threads span one node's 128 features
__global__ void gcn_finalize(const float4* __restrict__ agg,
                             const float* __restrict__ inorm,
                             const float4* __restrict__ bias,
                             float4* __restrict__ out, int total4) {
    int gid = blockIdx.x * blockDim.x + threadIdx.x;
    if (gid < total4) {
        int n  = gid >> 5;            // 32 float4 per node
        int f4 = gid & 31;
        float s  = inorm[n];
        float4 a = agg[gid];
        float4 b = bias[f4];
        float4 v;
        v.x = fmaxf(a.x * s + b.x, 0.f);
        v.y = fmaxf(a.y * s + b.y, 0.f);
        v.z = fmaxf(a.z * s + b.z, 0.f);
        v.w = fmaxf(a.w * s + b.w, 0.f);
        out[gid] = v;
    }
}

extern "C" void kernel_launch(void* const* d_in, const int* in_sizes, int n_in,
                              void* d_out, int out_size, void* d_ws, size_t ws_size,
                              hipStream_t stream) {
    const float* feat = (const float*)d_in[0];
    const int*   src  = (const int*)d_in[1];
    const int*   dst  = (const int*)d_in[2];
    const float* W1   = (const float*)d_in[3];
    const float* b1   = (const float*)d_in[4];
    const float* W2   = (const float*)d_in[5];
    const float* b2   = (const float*)d_in[6];
    float* out = (float*)d_out;

    const int N  = in_sizes[0] / HID;
    const int E  = in_sizes[1];
    const int NF = N * HID;               // 12.8M elements
    const int NF4 = NF / 4;

    // workspace layout (floats): degO[N] degI[N] t[N*128] agg[N*128] h1[N*128]
    float* ws   = (float*)d_ws;
    float* degO = ws;
    float* degI = ws + N;
    float* t    = ws + 2 * (size_t)N;
    float* agg  = t + (size_t)NF;
    float* h1   = agg + (size_t)NF;

    const int B = 256;
    const int edgeThreads = E * HID;      // 204.8M, fits int32

    // degrees -> norms (degO becomes out_norm, degI becomes in_norm, in place)
    // 2N is a multiple of 4 (N=100000)
    zero_f32v4<<<(2 * N / 4 + B - 1) / B, B, 0, stream>>>((float4*)degO, 2 * N / 4);
    degree_count<<<(E + B - 1) / B, B, 0, stream>>>(src, dst, (unsigned*)degO,
                                                    (unsigned*)degI, E);
    deg_to_norm<<<(2 * N + B - 1) / B, B, 0, stream>>>(degO, 2 * N);

    const int gBlocks = (N + TROWS - 1) / TROWS;

    // ---- layer 1
    zero_f32v4<<<(NF4 + B - 1) / B, B, 0, stream>>>((float4*)agg, NF4);
    gcn_transform_wmma<<<gBlocks, 128, 0, stream>>>(feat, degO, W1, t, N);
    gcn_scatter<<<(edgeThreads + B - 1) / B, B, 0, stream>>>(t, src, dst, agg, E);
    gcn_finalize<<<(NF4 + B - 1) / B, B, 0, stream>>>((const float4*)agg, degI,
                                                      (const float4*)b1, (float4*)h1, NF4);

    // ---- layer 2
    zero_f32v4<<<(NF4 + B - 1) / B, B, 0, stream>>>((float4*)agg, NF4);
    gcn_transform_wmma<<<gBlocks, 128, 0, stream>>>(h1, degO, W2, t, N);
    gcn_scatter<<<(edgeThreads + B - 1) / B, B, 0, stream>>>(t, src, dst, agg, E);
    gcn_finalize<<<(NF4 + B - 1) / B, B, 0, stream>>>((const float4*)agg, degI,
                                                      (const float4*)b2, (float4*)out, NF4);
}